// GAT_15994458211261
// MI455X (gfx1250) — compile-verified
//
#include <hip/hip_runtime.h>
#include <math.h>

// ---------------- problem constants ----------------
static constexpr int N    = 50000;
static constexpr int E    = 800000;
static constexpr int ETOT = E + N;      // with self loops
static constexpr int F    = 16;
static constexpr int ED   = 7;
static constexpr int HC   = 64;         // H*C = 4*16
static constexpr float NEG_ATT = 0.2f;
static constexpr float NEG     = 0.01f;
static constexpr int NEG_INF_BITS = 0xFF800000;   // -inf as float bits

typedef __attribute__((ext_vector_type(2))) float v2f;
typedef __attribute__((ext_vector_type(8))) float v8f;

static __device__ __forceinline__ void atomic_add_f(float* p, float v) {
    __hip_atomic_fetch_add(p, v, __ATOMIC_RELAXED, __HIP_MEMORY_SCOPE_AGENT);
}
// Ordered-float max via integer atomics (single HW instruction either path).
// Valid for any mix of signs when initialized to -inf bits.
static __device__ __forceinline__ void atomic_max_fbits(int* p, float v) {
    if (v >= 0.0f) atomicMax(p, __float_as_int(v));
    else           atomicMin((unsigned int*)p, __float_as_uint(v));
}

// ---------------- zero a float region ----------------
__global__ void k_zero(float* __restrict__ p, int n) {
    int i = blockIdx.x * blockDim.x + threadIdx.x;
    if (i < n) p[i] = 0.0f;
}

// ---------------- self-loop attr: scatter sums ----------------
__global__ void k_eattr_scatter(const float* __restrict__ eattr,
                                const int* __restrict__ ei,
                                float* __restrict__ lsum,   // [N*7]
                                float* __restrict__ cnt) {  // [N]
    int e = blockIdx.x * blockDim.x + threadIdx.x;
    if (e >= E) return;
    int d = ei[E + e];
    #pragma unroll
    for (int j = 0; j < ED; ++j)
        atomic_add_f(&lsum[d * ED + j], eattr[e * ED + j]);
    atomic_add_f(&cnt[d], 1.0f);
}

__global__ void k_loop_mean(float* __restrict__ lsum, const float* __restrict__ cnt) {
    int n = blockIdx.x * blockDim.x + threadIdx.x;
    if (n >= N) return;
    float inv = 1.0f / fmaxf(cnt[n], 1.0f);
    #pragma unroll
    for (int j = 0; j < ED; ++j) lsum[n * ED + j] *= inv;
}

// ---------------- WMMA GEMM: out{L,R} = A(NxK) @ W{l,r}(Kx64) + b ----------------
// block = 128 threads (4 waves); wave w -> 16x16 tile at cols [16w,16w+16),
// rows [16*blockIdx.x, +16).
template <int K>
__global__ void k_gemm_pair(const float* __restrict__ A,
                            const float* __restrict__ Wl, const float* __restrict__ bl,
                            const float* __restrict__ Wr, const float* __restrict__ br,
                            float* __restrict__ outL, float* __restrict__ outR) {
    const int wave = threadIdx.x >> 5;
    const int lane = threadIdx.x & 31;
    const int r    = lane & 15;
    const int hi   = lane >> 4;           // 0: K+0/1, 1: K+2/3
    const int row0 = blockIdx.x * 16;
    const int col0 = wave * 16;
    const int arow = row0 + r;
    const int bcol = col0 + r;

    v8f accL = {};
    v8f accR = {};
    #pragma unroll
    for (int k0 = 0; k0 < K; k0 += 4) {
        const int kk = k0 + hi * 2;
        v2f a = *reinterpret_cast<const v2f*>(A + arow * K + kk);   // 8B aligned
        v2f b;
        b.x = Wl[kk * HC + bcol];
        b.y = Wl[(kk + 1) * HC + bcol];
        accL = __builtin_amdgcn_wmma_f32_16x16x4_f32(false, a, false, b,
                                                     (short)0, accL, false, false);
        v2f b2;
        b2.x = Wr[kk * HC + bcol];
        b2.y = Wr[(kk + 1) * HC + bcol];
        accR = __builtin_amdgcn_wmma_f32_16x16x4_f32(false, a, false, b2,
                                                     (short)0, accR, false, false);
    }
    // C/D layout: VGPR i -> M = i (lanes 0-15) / i+8 (lanes 16-31); N = lane&15
    const float bLv = bl[bcol];
    const float bRv = br[bcol];
    #pragma unroll
    for (int i = 0; i < 8; ++i) {
        const int m = row0 + hi * 8 + i;
        outL[m * HC + bcol] = accL[i] + bLv;
        outR[m * HC + bcol] = accR[i] + bRv;
    }
}

// ---------------- init attention accumulators ----------------
__global__ void k_init_attn(float* __restrict__ acc,    // [N*64]
                            int*   __restrict__ amax,   // [N*4] float bits
                            float* __restrict__ z) {    // [N*4]
    int i = blockIdx.x * blockDim.x + threadIdx.x;
    if (i < N * HC) acc[i] = 0.0f;
    if (i < N * 4) { amax[i] = NEG_INF_BITS; z[i] = 0.0f; }
}

// ---------------- pass 1: logits + segment max ----------------
// one thread per (edge, head)
__global__ void k_logits(const float* __restrict__ xl, const float* __restrict__ xr,
                         const float* __restrict__ eattr,  // [E*7]
                         const float* __restrict__ lattr,  // [N*7] self-loop attrs
                         const int* __restrict__ ei,
                         const float* __restrict__ We,     // [7*64]
                         const float* __restrict__ att,    // [64]
                         float* __restrict__ alog,         // [ETOT*4]
                         int*   __restrict__ amax) {       // [N*4] float bits
    __shared__ float sWe[ED * HC];
    __shared__ float sAtt[HC];
    for (int i = threadIdx.x; i < ED * HC; i += blockDim.x) sWe[i] = We[i];
    for (int i = threadIdx.x; i < HC; i += blockDim.x) sAtt[i] = att[i];
    __syncthreads();

    int t = blockIdx.x * blockDim.x + threadIdx.x;
    if (t >= ETOT * 4) return;
    const int e = t >> 2;
    const int h = t & 3;

    int s, d;
    const float* ea;
    if (e < E) { s = ei[e]; d = ei[E + e]; ea = eattr + (size_t)e * ED; }
    else       { s = e - E; d = s;         ea = lattr + (size_t)(e - E) * ED; }

    float eat[ED];
    #pragma unroll
    for (int j = 0; j < ED; ++j) eat[j] = ea[j];

    const float4* xls = reinterpret_cast<const float4*>(xl + (size_t)s * HC + h * 16);
    const float4* xrd = reinterpret_cast<const float4*>(xr + (size_t)d * HC + h * 16);

    float logit = 0.0f;
    #pragma unroll
    for (int q = 0; q < 4; ++q) {
        float4 vl = xls[q];
        float4 vr = xrd[q];
        float xv[4] = { vl.x + vr.x, vl.y + vr.y, vl.z + vr.z, vl.w + vr.w };
        #pragma unroll
        for (int k = 0; k < 4; ++k) {
            const int c = q * 4 + k;
            float em = 0.0f;
            #pragma unroll
            for (int j = 0; j < ED; ++j) em = fmaf(eat[j], sWe[j * HC + h * 16 + c], em);
            float v = xv[k] + em;
            v = (v > 0.0f) ? v : NEG_ATT * v;          // leaky_relu(0.2)
            logit = fmaf(v, sAtt[h * 16 + c], logit);
        }
    }
    alog[(size_t)e * 4 + h] = logit;
    atomic_max_fbits(&amax[d * 4 + h], logit);
}

// ---------------- pass 2: exp, denominator, unnormalized scatter-sum ----------------
__global__ void k_accum(const float* __restrict__ xl,
                        const int* __restrict__ ei,
                        const float* __restrict__ alog,
                        const int*   __restrict__ amax,   // float bits
                        float* __restrict__ z,
                        float* __restrict__ acc) {
    int t = blockIdx.x * blockDim.x + threadIdx.x;
    if (t >= ETOT * 4) return;
    const int e = t >> 2;
    const int h = t & 3;
    int s, d;
    if (e < E) { s = ei[e]; d = ei[E + e]; }
    else       { s = e - E; d = s; }

    const float m  = __int_as_float(amax[d * 4 + h]);
    const float ea = __expf(alog[(size_t)e * 4 + h] - m);
    atomic_add_f(&z[d * 4 + h], ea);

    const float4* xs = reinterpret_cast<const float4*>(xl + (size_t)s * HC + h * 16);
    float* ad = acc + (size_t)d * HC + h * 16;
    #pragma unroll
    for (int i = 0; i < 4; ++i) {
        float4 v = xs[i];
        atomic_add_f(&ad[4 * i + 0], ea * v.x);
        atomic_add_f(&ad[4 * i + 1], ea * v.y);
        atomic_add_f(&ad[4 * i + 2], ea * v.z);
        atomic_add_f(&ad[4 * i + 3], ea * v.w);
    }
}

// ---------------- per-node finalize: normalize + bias + leaky(0.01) ----------------
__global__ void k_finalize(const float* __restrict__ acc,
                           const float* __restrict__ z,
                           const float* __restrict__ bo,
                           float* __restrict__ hout) {
    int i = blockIdx.x * blockDim.x + threadIdx.x;
    if (i >= N * HC) return;
    const int n = i >> 6;
    const int c = i & 63;
    float v = acc[i] / z[n * 4 + (c >> 4)] + bo[c];
    hout[i] = (v > 0.0f) ? v : NEG * v;
}

// ---------------- regression head: out[n] = h2[n,:] @ Ws + bs ----------------
__global__ void k_head(const float* __restrict__ h2,
                       const float* __restrict__ Ws,
                       const float* __restrict__ bs,
                       float* __restrict__ out) {
    int n = blockIdx.x * blockDim.x + threadIdx.x;
    if (n >= N) return;
    const float4* hv = reinterpret_cast<const float4*>(h2 + (size_t)n * HC);
    const float4* wv = reinterpret_cast<const float4*>(Ws);
    float s = 0.0f;
    #pragma unroll
    for (int j = 0; j < 16; ++j) {
        float4 a = hv[j], b = wv[j];
        s = fmaf(a.x, b.x, s); s = fmaf(a.y, b.y, s);
        s = fmaf(a.z, b.z, s); s = fmaf(a.w, b.w, s);
    }
    out[n] = s + bs[0];
}

static inline int blocks(long long n, int bs) { return (int)((n + bs - 1) / bs); }

extern "C" void kernel_launch(void* const* d_in, const int* in_sizes, int n_in,
                              void* d_out, int out_size, void* d_ws, size_t ws_size,
                              hipStream_t stream) {
    const float* x     = (const float*)d_in[0];
    const int*   ei    = (const int*)d_in[1];
    const float* eattr = (const float*)d_in[2];
    const float* Wl1 = (const float*)d_in[3];  const float* bl1 = (const float*)d_in[4];
    const float* Wr1 = (const float*)d_in[5];  const float* br1 = (const float*)d_in[6];
    const float* We1 = (const float*)d_in[7];  const float* att1 = (const float*)d_in[8];
    const float* bo1 = (const float*)d_in[9];
    const float* Wl2 = (const float*)d_in[10]; const float* bl2 = (const float*)d_in[11];
    const float* Wr2 = (const float*)d_in[12]; const float* br2 = (const float*)d_in[13];
    const float* We2 = (const float*)d_in[14]; const float* att2 = (const float*)d_in[15];
    const float* bo2 = (const float*)d_in[16];
    const float* Ws  = (const float*)d_in[17]; const float* bs = (const float*)d_in[18];
    float* out = (float*)d_out;

    // workspace layout (floats)
    float* ws = (float*)d_ws;
    size_t o = 0;
    float* lsum = ws + o; o += (size_t)N * ED;   // loop attr sum -> mean (in place)
    float* cnt  = ws + o; o += (size_t)N;
    float* xl   = ws + o; o += (size_t)N * HC;
    float* xr   = ws + o; o += (size_t)N * HC;
    float* alog = ws + o; o += (size_t)ETOT * 4;
    int*   amax = (int*)(ws + o); o += (size_t)N * 4;   // float bits
    float* zden = ws + o; o += (size_t)N * 4;
    float* acc  = ws + o; o += (size_t)N * HC;
    float* hbuf = ws + o; o += (size_t)N * HC;   // layer output / next layer input

    const int BT = 256;

    // ---- self-loop edge attr = mean of incoming edge attrs ----
    k_zero<<<blocks((long long)N * (ED + 1), BT), BT, 0, stream>>>(lsum, N * (ED + 1));
    k_eattr_scatter<<<blocks(E, BT), BT, 0, stream>>>(eattr, ei, lsum, cnt);
    k_loop_mean<<<blocks(N, BT), BT, 0, stream>>>(lsum, cnt);

    // ---- layer 1 ----
    k_gemm_pair<F><<<N / 16, 128, 0, stream>>>(x, Wl1, bl1, Wr1, br1, xl, xr);
    k_init_attn<<<blocks((long long)N * HC, BT), BT, 0, stream>>>(acc, amax, zden);
    k_logits<<<blocks((long long)ETOT * 4, BT), BT, 0, stream>>>(xl, xr, eattr, lsum, ei,
                                                                 We1, att1, alog, amax);
    k_accum<<<blocks((long long)ETOT * 4, BT), BT, 0, stream>>>(xl, ei, alog, amax, zden, acc);
    k_finalize<<<blocks((long long)N * HC, BT), BT, 0, stream>>>(acc, zden, bo1, hbuf);

    // ---- layer 2 ----
    k_gemm_pair<HC><<<N / 16, 128, 0, stream>>>(hbuf, Wl2, bl2, Wr2, br2, xl, xr);
    k_init_attn<<<blocks((long long)N * HC, BT), BT, 0, stream>>>(acc, amax, zden);
    k_logits<<<blocks((long long)ETOT * 4, BT), BT, 0, stream>>>(xl, xr, eattr, lsum, ei,
                                                                 We2, att2, alog, amax);
    k_accum<<<blocks((long long)ETOT * 4, BT), BT, 0, stream>>>(xl, ei, alog, amax, zden, acc);
    k_finalize<<<blocks((long long)N * HC, BT), BT, 0, stream>>>(acc, zden, bo2, hbuf);

    // ---- regression head ----
    k_head<<<blocks(N, BT), BT, 0, stream>>>(hbuf, Ws, bs, out);
}